// PCGen_14413910245827
// MI455X (gfx1250) — compile-verified
//
#include <hip/hip_runtime.h>
#include <hip/hip_bf16.h>
#include <math.h>

// ---------------------------------------------------------------------------
// PCGen pipeline for MI455X (gfx1250, wave32).
// Conv chain = batched GEMMs on V_WMMA_F32_16X16X4_F32 (fp32 matrix path,
// exact dtype of the reference). Activations are kept TRANSPOSED (B, M, C)
// between stages so that:
//   - B-fragments are contiguous b64 loads (row k = vgpr + 2*lanehalf),
//   - D-tiles store 8 consecutive channels per lane -> 2x global_store_b128.
// Each wave owns a 16(O) x 64(N) tile (4 accumulators) to amortize A loads:
// ~1.25 vmem loads per wmma. Graph filtering = LDS-staged brute-force kNN.
// ---------------------------------------------------------------------------

typedef __attribute__((ext_vector_type(2))) float v2f;
typedef __attribute__((ext_vector_type(8))) float v8f;

#define M_PTS 2048
#define B_SZ  32

__device__ __forceinline__ v8f wmma_f32_16x16x4(v2f a, v2f b, v8f c) {
  // 8 args: (neg_a, A, neg_b, B, c_mod, C, reuse_a, reuse_b)
  return __builtin_amdgcn_wmma_f32_16x16x4_f32(
      false, a, false, b, (short)0, c, false, false);
}

// ---------------------------------------------------------------------------
// Conv stage:  Y[b][m][o] = act( sum_c W[o][c] * X[b][.][.] + bias[o] )
//   XT = true : X is (B, M, CIN)  -> contiguous b64 B-frag loads
//   XT = false: X is (B, CIN, M)  -> strided b32 B-frag loads (stage 1 only)
// Output is always (B, M, Cout).
// EPI: 0 = relu, 1 = clip(-1,1) then * z[b][o].
// Block: 256 thr = 8 waves. Wave tile 16(O) x 64(N), block tile 64 x 128.
// Fragment addressing (ISA 7.12.2, 32-bit layouts):
//   A 16x4 : lane&15 = out row; K base = c + 2*(lane>>4); vgpr0=K, vgpr1=K+1
//   B 4x16 : lane&15 = col;     K row  = vgprIdx + 2*(lane>>4)
//   C/D    : row = o0 + j + 8*(lane>>4), col = lane&15
// ---------------------------------------------------------------------------
template <int CIN, int EPI, bool XT>
__global__ __launch_bounds__(256) void conv_gemm(
    const float* __restrict__ W, const float* __restrict__ bias,
    const float* __restrict__ X, float* __restrict__ Y,
    const float* __restrict__ zvec, int Cout)
{
  const int lane = threadIdx.x & 31;
  const int wave = threadIdx.x >> 5;
  const int hi   = lane >> 4;       // lane half selects K +2
  const int col  = lane & 15;
  const int b    = blockIdx.z;

  const int o0 = blockIdx.y * 64 + (wave >> 1) * 16;    // 4 waves along O
  const int nb = blockIdx.x * 128 + (wave & 1) * 64;    // 2 waves along N, 64 cols each

  // A fragment source: W row (o0+col), K offset 2*hi (contiguous pair)
  const float* __restrict__ wp = W + (size_t)(o0 + col) * CIN + 2 * hi;

  v8f acc0 = {}, acc1 = {}, acc2 = {}, acc3 = {};

  if (XT) {
    // X: (B, M, CIN); per lane: column n = nb + t*16 + col, contiguous K pair
    const float* __restrict__ xb =
        X + ((size_t)b * M_PTS + nb + col) * CIN + 2 * hi;
#pragma unroll 4
    for (int c = 0; c < CIN; c += 4) {
      const v2f a  = *(const v2f*)(wp + c);
      const v2f f0 = *(const v2f*)(xb + c);
      const v2f f1 = *(const v2f*)(xb + (size_t)16 * CIN + c);
      const v2f f2 = *(const v2f*)(xb + (size_t)32 * CIN + c);
      const v2f f3 = *(const v2f*)(xb + (size_t)48 * CIN + c);
      acc0 = wmma_f32_16x16x4(a, f0, acc0);
      acc1 = wmma_f32_16x16x4(a, f1, acc1);
      acc2 = wmma_f32_16x16x4(a, f2, acc2);
      acc3 = wmma_f32_16x16x4(a, f3, acc3);
    }
  } else {
    // X: (B, CIN, M); B-frag rows are M apart
    const float* __restrict__ xb =
        X + (size_t)b * CIN * M_PTS + (size_t)(2 * hi) * M_PTS + (nb + col);
#pragma unroll
    for (int c = 0; c < CIN; c += 4) {
      const v2f a = *(const v2f*)(wp + c);
      const float* r0 = xb + (size_t)c * M_PTS;
      const float* r1 = r0 + M_PTS;
      v2f f0; f0.x = r0[0];  f0.y = r1[0];
      v2f f1; f1.x = r0[16]; f1.y = r1[16];
      v2f f2; f2.x = r0[32]; f2.y = r1[32];
      v2f f3; f3.x = r0[48]; f3.y = r1[48];
      acc0 = wmma_f32_16x16x4(a, f0, acc0);
      acc1 = wmma_f32_16x16x4(a, f1, acc1);
      acc2 = wmma_f32_16x16x4(a, f2, acc2);
      acc3 = wmma_f32_16x16x4(a, f3, acc3);
    }
  }

  // Epilogue: per lane, 8 consecutive channels o = o0 + 8*hi + (0..7)
  const int ob = o0 + 8 * hi;
  const float4 bLo = *(const float4*)(bias + ob);
  const float4 bHi = *(const float4*)(bias + ob + 4);
  float4 zLo = {1.f, 1.f, 1.f, 1.f}, zHi = {1.f, 1.f, 1.f, 1.f};
  if (EPI == 1) {
    zLo = *(const float4*)(zvec + b * 512 + ob);
    zHi = *(const float4*)(zvec + b * 512 + ob + 4);
  }

  v8f accs[4] = {acc0, acc1, acc2, acc3};
#pragma unroll
  for (int t = 0; t < 4; ++t) {
    const int n = nb + t * 16 + col;
    float v[8];
#pragma unroll
    for (int j = 0; j < 8; ++j) {
      float x = accs[t][j] + ((j < 4) ? (&bLo.x)[j] : (&bHi.x)[j - 4]);
      if (EPI == 0) {
        x = fmaxf(x, 0.0f);
      } else {
        x = fminf(fmaxf(x, -1.0f), 1.0f);
        x *= (j < 4) ? (&zLo.x)[j] : (&zHi.x)[j - 4];
      }
      v[j] = x;
    }
    float* __restrict__ yp = Y + ((size_t)b * M_PTS + n) * Cout + ob;
    *(float4*)(yp)     = make_float4(v[0], v[1], v[2], v[3]);
    *(float4*)(yp + 4) = make_float4(v[4], v[5], v[6], v[7]);
  }
}

// ---------------------------------------------------------------------------
// Stage 7: 64 -> 3 channels (too thin for WMMA). One thread per (b, m).
// Input is transposed (B, M, 64) -> fully contiguous float4 stream per thread.
// Output x7 is (B, 3, M) for the graph kernel.
// ---------------------------------------------------------------------------
__global__ __launch_bounds__(256) void conv_final(
    const float* __restrict__ W7, const float* __restrict__ b7,
    const float* __restrict__ X, float* __restrict__ Y)
{
  const int idx = blockIdx.x * 256 + threadIdx.x;   // 0 .. B*M-1
  const int b = idx / M_PTS;
  const int m = idx % M_PTS;

  const float4* __restrict__ x4 = (const float4*)(X + ((size_t)b * M_PTS + m) * 64);
  float a0 = b7[0], a1 = b7[1], a2 = b7[2];
#pragma unroll
  for (int i = 0; i < 16; ++i) {
    const float4 v = x4[i];
    const int c = 4 * i;
    a0 = fmaf(W7[c], v.x, a0);   a0 = fmaf(W7[c + 1], v.y, a0);
    a0 = fmaf(W7[c + 2], v.z, a0); a0 = fmaf(W7[c + 3], v.w, a0);
    a1 = fmaf(W7[64 + c], v.x, a1);   a1 = fmaf(W7[64 + c + 1], v.y, a1);
    a1 = fmaf(W7[64 + c + 2], v.z, a1); a1 = fmaf(W7[64 + c + 3], v.w, a1);
    a2 = fmaf(W7[128 + c], v.x, a2);   a2 = fmaf(W7[128 + c + 1], v.y, a2);
    a2 = fmaf(W7[128 + c + 2], v.z, a2); a2 = fmaf(W7[128 + c + 3], v.w, a2);
  }
  float* __restrict__ yp = Y + (size_t)b * 3 * M_PTS + m;
  yp[0] = a0;
  yp[M_PTS] = a1;
  yp[2 * M_PTS] = a2;
}

// ---------------------------------------------------------------------------
// Graph filtering: per (batch, 256-point chunk). All 2048 pts of the batch in
// LDS (24 KB). Brute-force 4-NN with index-tracked sorted insert (strict '<'
// matches JAX top-k lowest-index tie-break; rank0 = self, dropped like the
// reference). softmax(-d/sigma) over 3 neighbors, out = 1.5*x - 0.5*agg.
// ---------------------------------------------------------------------------
__global__ __launch_bounds__(256) void graph_filter(
    const float* __restrict__ X, float* __restrict__ Out)
{
  __shared__ float pts[3 * M_PTS];   // [c][n]

  const int b = blockIdx.y;
  const int tid = threadIdx.x;
  const float* __restrict__ xb = X + (size_t)b * 3 * M_PTS;

  for (int i = tid; i < 3 * M_PTS; i += 256) pts[i] = xb[i];
  __syncthreads();

  const int m = blockIdx.x * 256 + tid;
  const float px = pts[m];
  const float py = pts[M_PTS + m];
  const float pz = pts[2 * M_PTS + m];

  float d0 = 3.4e38f, d1 = 3.4e38f, d2 = 3.4e38f, d3 = 3.4e38f;
  int i0 = 0, i1 = 0, i2 = 0, i3 = 0;
  for (int n = 0; n < M_PTS; ++n) {
    const float dx = pts[n] - px;
    const float dy = pts[M_PTS + n] - py;
    const float dz = pts[2 * M_PTS + n] - pz;
    const float dd = dx * dx + dy * dy + dz * dz;
    if (dd < d3) {
      if (dd < d0) {
        d3 = d2; i3 = i2; d2 = d1; i2 = i1; d1 = d0; i1 = i0; d0 = dd; i0 = n;
      } else if (dd < d1) {
        d3 = d2; i3 = i2; d2 = d1; i2 = i1; d1 = dd; i1 = n;
      } else if (dd < d2) {
        d3 = d2; i3 = i2; d2 = dd; i2 = n;
      } else {
        d3 = dd; i3 = n;
      }
    }
  }

  const float e1 = fmaxf(d1, 0.0f);
  const float e2 = fmaxf(d2, 0.0f);
  const float e3 = fmaxf(d3, 0.0f);
  const float sigma = sqrtf((e1 + e2 + e3) * (1.0f / 3.0f));
  const float inv = 1.0f / sigma;
  const float a1 = -e1 * inv, a2 = -e2 * inv, a3 = -e3 * inv;
  const float mx = fmaxf(a1, fmaxf(a2, a3));
  float w1 = expf(a1 - mx), w2 = expf(a2 - mx), w3 = expf(a3 - mx);
  const float r = 1.0f / (w1 + w2 + w3);
  w1 *= r; w2 *= r; w3 *= r;

  float* __restrict__ out = Out + (size_t)b * 3 * M_PTS + m;
#pragma unroll
  for (int c = 0; c < 3; ++c) {
    const float* pc = pts + c * M_PTS;
    const float agg = w1 * pc[i1] + w2 * pc[i2] + w3 * pc[i3];
    out[(size_t)c * M_PTS] = 1.5f * pc[m] - 0.5f * agg;
  }
}

// ---------------------------------------------------------------------------
// Launch. Inputs (setup_inputs order):
//  0:z 1:s 2:w1 3:b1 4:w2 5:b2 6:w3 7:b3 8:w4 9:b4 10:w5 11:b5 12:w6 13:b6
//  14:w7 15:b7
// ---------------------------------------------------------------------------
extern "C" void kernel_launch(void* const* d_in, const int* in_sizes, int n_in,
                              void* d_out, int out_size, void* d_ws, size_t ws_size,
                              hipStream_t stream) {
  const float* z  = (const float*)d_in[0];
  const float* s  = (const float*)d_in[1];
  const float* w1 = (const float*)d_in[2];   const float* b1 = (const float*)d_in[3];
  const float* w2 = (const float*)d_in[4];   const float* b2 = (const float*)d_in[5];
  const float* w3 = (const float*)d_in[6];   const float* b3 = (const float*)d_in[7];
  const float* w4 = (const float*)d_in[8];   const float* b4 = (const float*)d_in[9];
  const float* w5 = (const float*)d_in[10];  const float* b5 = (const float*)d_in[11];
  const float* w6 = (const float*)d_in[12];  const float* b6 = (const float*)d_in[13];
  const float* w7 = (const float*)d_in[14];  const float* b7 = (const float*)d_in[15];
  float* out = (float*)d_out;

  // Ping-pong activation buffers (max stage = 32*2048*512 fp32 = 128 MB each).
  const size_t bufElems = (size_t)B_SZ * M_PTS * 512;
  float* bufA = (float*)d_ws;
  float* bufB = bufA + bufElems;
  float* x7   = bufB + bufElems;              // 32*3*2048 floats

  dim3 blk(256);
  const dim3 gN16(M_PTS / 128, 0, 0);  // helper: 16 blocks along N
  // stage 1: 16 -> 256, relu           (s is (B,16,M): XT=false)
  conv_gemm<16, 0, false><<<dim3(16, 4, 32), blk, 0, stream>>>(w1, b1, s, bufA, nullptr, 256);
  // stage 2: 256 -> 512, clip then *z
  conv_gemm<256, 1, true><<<dim3(16, 8, 32), blk, 0, stream>>>(w2, b2, bufA, bufB, z, 512);
  // stage 3: 512 -> 512, relu
  conv_gemm<512, 0, true><<<dim3(16, 8, 32), blk, 0, stream>>>(w3, b3, bufB, bufA, nullptr, 512);
  // stage 4: 512 -> 256, relu
  conv_gemm<512, 0, true><<<dim3(16, 4, 32), blk, 0, stream>>>(w4, b4, bufA, bufB, nullptr, 256);
  // stage 5: 256 -> 128, relu
  conv_gemm<256, 0, true><<<dim3(16, 2, 32), blk, 0, stream>>>(w5, b5, bufB, bufA, nullptr, 128);
  // stage 6: 128 -> 64, relu
  conv_gemm<128, 0, true><<<dim3(16, 1, 32), blk, 0, stream>>>(w6, b6, bufA, bufB, nullptr, 64);
  // stage 7: 64 -> 3 (reads transposed, writes (B,3,M))
  conv_final<<<dim3((B_SZ * M_PTS) / 256), blk, 0, stream>>>(w7, b7, bufB, x7);
  // graph filtering
  graph_filter<<<dim3(M_PTS / 256, B_SZ), blk, 0, stream>>>(x7, out);
}